// Attention_6201932775850
// MI455X (gfx1250) — compile-verified
//
#include <hip/hip_runtime.h>
#include <math.h>

typedef __attribute__((ext_vector_type(2))) float v2f;
typedef __attribute__((ext_vector_type(8))) float v8f;

#define B_   16
#define H_   32
#define KH_  8
#define G_   4
#define D_   128
#define S_   8192
#define NPART 16
#define KEYS_PER_PART (S_ / NPART)   // 512
#define TILES (KEYS_PER_PART / 16)   // 32
#define SCALE_ 0.08838834764831845f
#define WS_STRIDE (8 + G_ * D_)      // floats per partial record: m[4], l[4], acc[4][128]

// ---------------------------------------------------------------------------
// Kernel 1: one wave32 per (b, kv_head, partition). Flash-decoding partial.
// All global loads for a tile are batch-issued ahead of the WMMA chains so
// the hardware keeps many loads in flight (counted s_wait_loadcnt instead of
// load->wait(0)->wmma serialization).
// ---------------------------------------------------------------------------
__global__ __launch_bounds__(32)
void attn_partial(const float* __restrict__ q, const float* __restrict__ k,
                  const float* __restrict__ v, const float* __restrict__ kc,
                  const float* __restrict__ vc, const long long* __restrict__ slot,
                  float* __restrict__ ws) {
  const int part = blockIdx.x & (NPART - 1);
  const int bk   = blockIdx.x / NPART;
  const int kh   = bk & (KH_ - 1);
  const int b    = bk / KH_;
  const int lane = threadIdx.x;
  const int half = lane >> 4;     // which 16-lane half of the wave
  const int l15  = lane & 15;
  const int geff = l15 & 3;       // query row (replicated across lane groups)

  __shared__ __align__(16) float q_s[G_][132]; // padded: stride 132 % 64 != 0
  __shared__ __align__(16) float p_s[16][20];  // softmax weights [key][qrow]

  // Q rows for this kv head are 512 contiguous floats: q[b, kh*G .. kh*G+3, :]
  const float* qbase = q + ((size_t)b * H_ + (size_t)kh * G_) * D_;
  for (int i = lane; i < G_ * D_; i += 32) q_s[i >> 7][i & 127] = qbase[i];
  __syncthreads();

  const long long slt = slot[b];
  const size_t sstride = (size_t)KH_ * D_;          // floats between seq steps
  const float* kc_b = kc + (size_t)b * S_ * sstride + (size_t)kh * D_;
  const float* vc_b = vc + (size_t)b * S_ * sstride + (size_t)kh * D_;
  const float* knew = k + ((size_t)b * KH_ + kh) * D_;
  const float* vnew = v + ((size_t)b * KH_ + kh) * D_;

  float m = -INFINITY, l = 0.f;
  v8f acc[8];                         // O^T accumulators: 8 d-tiles of 16
  #pragma unroll
  for (int i = 0; i < 8; ++i) acc[i] = v8f{};

  const int s0 = part * KEYS_PER_PART;
  for (int t = 0; t < TILES; ++t) {
    const int key_base = s0 + t * 16;

    // ---- per-lane effective K row (slot substitution folded into pointer)
    const int mykey = key_base + l15;     // A-matrix row = key for this lane
    const float* krow = (mykey == slt) ? knew : (kc_b + (size_t)mykey * sstride);

    if (t + 1 < TILES) {                  // CDNA5 global_prefetch_b8
      const int nkey = mykey + 16;
      const float* npf = (nkey == slt) ? knew : (kc_b + (size_t)nkey * sstride);
      __builtin_prefetch((const void*)npf, 0, 0);
    }

    // ---- batch-issue ALL K fragment loads for this tile (32x b64)
    v2f ka[32];
    const v2f* krow2 = (const v2f*)krow;
    #pragma unroll
    for (int s = 0; s < 32; ++s) ka[s] = krow2[2 * s + half];

    // ---- batch-issue first PV k-step's V fragment loads (independent of softmax)
    v2f va[2][8];
    {
      const int key0 = key_base + half * 2;
      const float* vrow0 = (key0     == slt) ? vnew : (vc_b + (size_t)key0 * sstride);
      const float* vrow1 = (key0 + 1 == slt) ? vnew : (vc_b + (size_t)(key0 + 1) * sstride);
      #pragma unroll
      for (int dt = 0; dt < 8; ++dt) {
        const int dcol = dt * 16 + l15;
        v2f a; a.x = vrow0[dcol]; a.y = vrow1[dcol];
        va[0][dt] = a;
      }
    }

    // ---- scores^T: C[key][g] += K_tile[16 x 4] * Q^T[4 x 16], over D=128
    // two partial accumulators break the C-register RAW chain
    v8f cs0 = v8f{}, cs1 = v8f{};
    const v2f* qrow2 = (const v2f*)(&q_s[geff][0]);
    #pragma unroll
    for (int s = 0; s < 32; s += 2) {
      v2f bq0 = qrow2[2 * s + half];
      v2f bq1 = qrow2[2 * (s + 1) + half];
      cs0 = __builtin_amdgcn_wmma_f32_16x16x4_f32(false, ka[s],     false, bq0,
                                                  (short)0, cs0, false, false);
      cs1 = __builtin_amdgcn_wmma_f32_16x16x4_f32(false, ka[s + 1], false, bq1,
                                                  (short)0, cs1, false, false);
    }
    v8f cs = cs0 + cs1;

    // ---- online softmax: lane holds 8 keys (half*8 + j) of query row geff
    float p[8];
    float smax = -INFINITY;
    #pragma unroll
    for (int j = 0; j < 8; ++j) { p[j] = cs[j] * SCALE_; smax = fmaxf(smax, p[j]); }
    smax = fmaxf(smax, __shfl_xor(smax, 16, 32));   // combine both key halves
    const float mnew = fmaxf(m, smax);
    const float r = __expf(m - mnew);               // exp(-inf)=0 on first tile
    float psum = 0.f;
    #pragma unroll
    for (int j = 0; j < 8; ++j) { p[j] = __expf(p[j] - mnew); psum += p[j]; }
    psum += __shfl_xor(psum, 16, 32);
    l = l * r + psum;
    m = mnew;
    if (l15 < 4) {
      #pragma unroll
      for (int j = 0; j < 8; ++j) p_s[half * 8 + j][l15] = p[j];
    }
    __syncthreads();

    // rescale running output (factor is uniform per lane -> v_pk_mul_f32)
    #pragma unroll
    for (int dt = 0; dt < 8; ++dt) acc[dt] *= r;

    // ---- O^T += V^T[16 x 4] * W^T[4 x 16]; double-buffered V fragments
    #pragma unroll
    for (int kk = 0; kk < 4; ++kk) {
      if (kk < 3) {                       // issue next k-step's loads now
        const int key0 = key_base + (kk + 1) * 4 + half * 2;
        const float* vrow0 = (key0     == slt) ? vnew : (vc_b + (size_t)key0 * sstride);
        const float* vrow1 = (key0 + 1 == slt) ? vnew : (vc_b + (size_t)(key0 + 1) * sstride);
        #pragma unroll
        for (int dt = 0; dt < 8; ++dt) {
          const int dcol = dt * 16 + l15;
          v2f a; a.x = vrow0[dcol]; a.y = vrow1[dcol];
          va[(kk + 1) & 1][dt] = a;
        }
      }
      v2f bw; bw.x = p_s[kk * 4 + half * 2][geff];
              bw.y = p_s[kk * 4 + half * 2 + 1][geff];
      #pragma unroll
      for (int dt = 0; dt < 8; ++dt) {
        acc[dt] = __builtin_amdgcn_wmma_f32_16x16x4_f32(false, va[kk & 1][dt], false, bw,
                                                        (short)0, acc[dt], false, false);
      }
    }
    __syncthreads();
  }

  // ---- write partial record: m[g], l[g], acc[g][d] (unnormalized)
  float* wsp = ws + (size_t)blockIdx.x * WS_STRIDE;
  if (l15 < 4) {
    if (half == 0) { wsp[l15] = m; wsp[4 + l15] = l; }
    float* dst = wsp + 8 + (size_t)l15 * D_;
    #pragma unroll
    for (int dt = 0; dt < 8; ++dt) {
      #pragma unroll
      for (int j = 0; j < 8; ++j)
        dst[dt * 16 + half * 8 + j] = acc[dt][j];   // C: vgpr j -> row j + 8*half
    }
  }
}

// ---------------------------------------------------------------------------
// Kernel 2: merge NPART partials per (b, h) with log-sum-exp combining.
// ---------------------------------------------------------------------------
__global__ __launch_bounds__(128)
void attn_reduce(const float* __restrict__ ws, float* __restrict__ out) {
  const int bh = blockIdx.x;            // [0, B*H)
  const int d  = threadIdx.x;           // [0, 128)
  const int b  = bh / H_;
  const int h  = bh % H_;
  const int kh = h / G_;
  const int g  = h % G_;
  const float* base = ws + ((size_t)(b * KH_ + kh) * NPART) * WS_STRIDE;

  float mstar = -INFINITY;
  #pragma unroll
  for (int p = 0; p < NPART; ++p)
    mstar = fmaxf(mstar, base[(size_t)p * WS_STRIDE + g]);

  float lstar = 0.f, o = 0.f;
  #pragma unroll 4
  for (int p = 0; p < NPART; ++p) {
    const float* rec = base + (size_t)p * WS_STRIDE;
    const float w = __expf(rec[g] - mstar);
    lstar += rec[4 + g] * w;
    o     += rec[8 + (size_t)g * D_ + d] * w;
  }
  out[(size_t)bh * D_ + d] = o / lstar;
}

extern "C" void kernel_launch(void* const* d_in, const int* in_sizes, int n_in,
                              void* d_out, int out_size, void* d_ws, size_t ws_size,
                              hipStream_t stream) {
  (void)in_sizes; (void)n_in; (void)out_size; (void)ws_size;
  const float*     q    = (const float*)d_in[0];
  const float*     k    = (const float*)d_in[1];
  const float*     v    = (const float*)d_in[2];
  const float*     kc   = (const float*)d_in[3];
  const float*     vc   = (const float*)d_in[4];
  const long long* slot = (const long long*)d_in[5];
  float* out = (float*)d_out;
  float* ws  = (float*)d_ws;   // needs 16*8*16*520*4 B ~= 4.1 MB

  attn_partial<<<dim3(B_ * KH_ * NPART), 32, 0, stream>>>(q, k, v, kc, vc, slot, ws);
  attn_reduce <<<dim3(B_ * H_),          128, 0, stream>>>(ws, out);
}